// RRNNCell_89867895702089
// MI455X (gfx1250) — compile-verified
//
#include <hip/hip_runtime.h>
#include <hip/hip_bf16.h>

// RRNN coupled 2-state scan, MI455X (gfx1250), wave32.
// Bandwidth-bound: 256 MB read-once + 134 MB write-once ~= 390 MB @ 23.3 TB/s
// => ~17 us floor; arithmetic is negligible, so the win is pure streaming:
// triple-buffered async global->LDS staging (ASYNCcnt, 32 b128 loads in
// flight per wave = 16 KB latency coverage), serial fp32 FMA recurrence from
// LDS, non-temporal hints on both streams (zero reuse anywhere).

constexpr int L      = 1024;         // sequence length
constexpr int B      = 32;           // batch
constexpr int D      = 512;          // channels per dir (BIDIR = 1)
constexpr int CH     = B * D;        // 16384 independent recurrence channels
constexpr int T      = 16;           // timesteps staged per chunk
constexpr int NBUF   = 3;            // triple buffer: 2-chunk-ahead prefetch
constexpr int LANES  = 32;           // wave32: one wave per block, no barriers
constexpr int NCHUNK = L / T;        // 64

__global__ __launch_bounds__(LANES)
void rrnn_scan_kernel(const float* __restrict__ u,
                      const float* __restrict__ c1_init,
                      const float* __restrict__ c2_init,
                      const float* __restrict__ eps,
                      float* __restrict__ out)
{
    // Staging tiles: [buffer][timestep][lane] of (u1,u2,f1,f2) = 24 KB
    __shared__ float4 buf[NBUF][T][LANES];

    const int lane = threadIdx.x;                 // 0..31
    const int c    = blockIdx.x * LANES + lane;   // channel = b*D + d, 0..16383
    const int d    = c & (D - 1);

    const float4* __restrict__ u4 = reinterpret_cast<const float4*>(u);

    float* __restrict__ out_c1  = out;                        // L*CH
    float* __restrict__ out_c2  = out + (size_t)L * CH;       // L*CH
    float* __restrict__ out_c1f = out + 2 * (size_t)L * CH;   // CH
    float* __restrict__ out_c2f = out_c1f + CH;               // CH

    const float e = eps[d];
    float c1 = c1_init[c];
    float c2 = c2_init[c];

    // Issue T async b128 loads: u[t0..t0+T-1, channel c] -> buf[p][*][lane].
    // Generic->u32 truncation of a __shared__ address yields the raw LDS
    // offset (LDS aperture lives in addr[63:32]; addr[31:0] is the offset).
    auto issue_chunk = [&](int chunk, int p) {
        unsigned long long gaddr =
            (unsigned long long)(u4 + (size_t)chunk * T * CH + c);
        unsigned laddr = (unsigned)(size_t)&buf[p][0][lane];
        #pragma unroll
        for (int tt = 0; tt < T; ++tt) {
            asm volatile("global_load_async_to_lds_b128 %0, %1, off th:TH_LOAD_NT"
                         :: "v"(laddr), "v"(gaddr)
                         : "memory");
            gaddr += (unsigned long long)CH * sizeof(float4);
            laddr += LANES * sizeof(float4);
        }
    };

    issue_chunk(0, 0);
    issue_chunk(1, 1);

    int pc = 0;         // buffer holding chunk k (compute)
    int pf = 2;         // buffer to fill with chunk k+2
    for (int k = 0; k < NCHUNK; ++k) {
        if (k + 2 < NCHUNK) {
            issue_chunk(k + 2, pf);
            // 3 chunks (48 loads) outstanding; in-order completion =>
            // <=32 outstanding means chunk k has fully landed.
            asm volatile("s_wait_asynccnt %0" :: "n"(2 * T) : "memory");
        } else if (k + 1 < NCHUNK) {
            asm volatile("s_wait_asynccnt %0" :: "n"(T) : "memory");
        } else {
            asm volatile("s_wait_asynccnt 0" ::: "memory");
        }

        const int t0 = k * T;
        #pragma unroll
        for (int tt = 0; tt < T; ++tt) {
            const float4 v = buf[pc][tt][lane];  // ds_load_b128: u1,u2,f1,f2
            const float c1n = fmaf(c1, v.z, v.x);             // c1*f1 + u1
            const float c2n = fmaf(c2, v.w, (e + c1) * v.y);  // c2*f2 + (e+c1_old)*u2
            c1 = c1n;
            c2 = c2n;
            const size_t idx = (size_t)(t0 + tt) * CH + c;
            __builtin_nontemporal_store(c1n, out_c1 + idx);   // write-once stream:
            __builtin_nontemporal_store(c2n, out_c2 + idx);   // keep GL2 clean
        }

        pc = (pc == NBUF - 1) ? 0 : pc + 1;
        pf = (pf == NBUF - 1) ? 0 : pf + 1;
    }

    out_c1f[c] = c1;
    out_c2f[c] = c2;
}

extern "C" void kernel_launch(void* const* d_in, const int* in_sizes, int n_in,
                              void* d_out, int out_size, void* d_ws, size_t ws_size,
                              hipStream_t stream) {
    const float* u       = (const float*)d_in[0];
    const float* c1_init = (const float*)d_in[1];
    const float* c2_init = (const float*)d_in[2];
    const float* eps     = (const float*)d_in[3];
    float* out           = (float*)d_out;

    dim3 grid(CH / LANES);   // 512 single-wave workgroups
    dim3 block(LANES);       // 32 threads = 1 wave32
    hipLaunchKernelGGL(rrnn_scan_kernel, grid, block, 0, stream,
                       u, c1_init, c2_init, eps, out);
}